// SequenceAttention_68848325755053
// MI455X (gfx1250) — compile-verified
//
#include <hip/hip_runtime.h>
#include <hip/hip_bf16.h>
#include <math.h>

// Additive attention, fused single kernel for MI455X (gfx1250, wave32).
// B=8, QS=64, KS=512, KD=512, H=8, dh=64, A=64.

typedef __attribute__((ext_vector_type(2))) float v2f;
typedef __attribute__((ext_vector_type(8))) float v8f;

#define BB   8
#define QSZ  64
#define KSZ  512
#define KDZ  512
#define HH   8
#define DH   64
#define AA   64
#define QT   16      // q rows per block
#define KTL  64      // k tile length
#define QPS  66      // qp LDS row stride (pad for banks)
#define KPS  66      // kp LDS row stride
#define SPS  516     // scores LDS row stride (even, +4 -> 4-bank row step)

__device__ __forceinline__ v8f wmma4(v2f a, v2f b, v8f c) {
  // D = A(16x4, f32) * B(4x16, f32) + C(16x16, f32)
  return __builtin_amdgcn_wmma_f32_16x16x4_f32(
      /*neg_a=*/false, a, /*neg_b=*/false, b,
      /*c_mod=*/(short)0, c, /*reuse_a=*/false, /*reuse_b=*/false);
}

__global__ __launch_bounds__(256)
void addatten_fused(const float* __restrict__ x,      // (B, KS, KD)
                    const float* __restrict__ query,  // (B, QS, QD)
                    const float* __restrict__ W,      // (A, 2*dh)  [Wk | Wq]
                    const float* __restrict__ bias,   // (A,)
                    const float* __restrict__ w2,     // (A,)
                    const unsigned char* __restrict__ mask, // (B, KS) bool
                    float* __restrict__ out)          // (B, QS, KD)
{
  __shared__ float qpb[QT * QPS];   // qp + bias for this q-chunk
  __shared__ float kpt[KTL * KPS];  // kp tile (current 64 k's)
  __shared__ float sc [QT * SPS];   // scores -> probabilities
  __shared__ float w2s[AA];

  const int blk = blockIdx.x;        // b*32 + h*4 + qc
  const int qc  = blk & 3;
  const int h   = (blk >> 2) & 7;
  const int b   = blk >> 5;
  const int q0  = qc * QT;

  const int tid  = threadIdx.x;
  const int lane = tid & 31;
  const int wv   = tid >> 5;       // wave id 0..7
  const int hl   = lane >> 4;      // lane half (selects K pair for A/B frags)
  const int ln   = lane & 15;      // M (A frag) or N (B/C/D frags)

  if (tid < AA) w2s[tid] = w2[tid];

  // ---------- Phase 1: qpb[q][a] = (qh(16x64) @ Wq^T)[q][a] + bias[a] ------
  if (wv < 4) {
    const int nt = wv;                               // a-tile (N)
    v8f c = {0.f,0.f,0.f,0.f,0.f,0.f,0.f,0.f};
    const float* qrow = query + ((size_t)b * QSZ + (q0 + ln)) * KDZ + h * DH;
    const float* wrow = W + (size_t)(nt * 16 + ln) * (2 * DH) + DH;  // Wq row
    #pragma unroll 4
    for (int kk = 0; kk < DH / 4; ++kk) {
      const int kb = kk * 4 + hl * 2;
      v2f av, bv;
      av.x = qrow[kb];  av.y = qrow[kb + 1];
      bv.x = wrow[kb];  bv.y = wrow[kb + 1];
      c = wmma4(av, bv, c);
    }
    const float bval = bias[nt * 16 + ln];
    #pragma unroll
    for (int r = 0; r < 8; ++r)
      qpb[(r + 8 * hl) * QPS + nt * 16 + ln] = c[r] + bval;
  }
  __syncthreads();

  // ---------- Phase 2: per k-tile: kp tile GEMM, then tanh/score ----------
  for (int kt = 0; kt < KSZ / KTL; ++kt) {
    const int kt0 = kt * KTL;

    // 2a: kpt(64x64) = kh(64x64) @ Wk^T   -- 16 tiles, 2 per wave
    #pragma unroll
    for (int j = 0; j < 2; ++j) {
      const int tt = wv * 2 + j;
      const int mt = tt >> 2, nt = tt & 3;
      v8f c = {0.f,0.f,0.f,0.f,0.f,0.f,0.f,0.f};
      const float* xrow = x + ((size_t)b * KSZ + (kt0 + mt * 16 + ln)) * KDZ + h * DH;
      const float* wrow = W + (size_t)(nt * 16 + ln) * (2 * DH);     // Wk row
      #pragma unroll 4
      for (int kk = 0; kk < DH / 4; ++kk) {
        const int kb = kk * 4 + hl * 2;
        v2f av, bv;
        av.x = xrow[kb];  av.y = xrow[kb + 1];
        bv.x = wrow[kb];  bv.y = wrow[kb + 1];
        c = wmma4(av, bv, c);
      }
      #pragma unroll
      for (int r = 0; r < 8; ++r)
        kpt[(mt * 16 + r + 8 * hl) * KPS + nt * 16 + ln] = c[r];
    }
    __syncthreads();

    // 2b: scores[q][kt0+k] = sum_a w2[a] * tanh(kpt[k][a] + qpb[q][a])
    for (int p = tid; p < QT * KTL; p += 256) {
      const int q = p >> 6;          // p / 64
      const int k = p & 63;
      const float* kr = &kpt[k * KPS];
      const float* qr = &qpb[q * QPS];
      float s = 0.f;
      #pragma unroll 8
      for (int a = 0; a < AA; ++a)
        s += w2s[a] * tanhf(kr[a] + qr[a]);
      if (mask[(size_t)b * KSZ + kt0 + k]) s = -INFINITY;
      sc[q * SPS + kt0 + k] = s;
    }
    __syncthreads();   // protect kpt before next tile overwrites it
  }

  // ---------- Phase 3: row softmax over k (512), in place ------------------
  #pragma unroll
  for (int rr = 0; rr < 2; ++rr) {
    const int q = wv * 2 + rr;
    float m = -INFINITY;
    for (int j = lane; j < KSZ; j += 32) m = fmaxf(m, sc[q * SPS + j]);
    for (int o = 16; o > 0; o >>= 1) m = fmaxf(m, __shfl_xor(m, o, 32));
    float s = 0.f;
    for (int j = lane; j < KSZ; j += 32) {
      const float e = expf(sc[q * SPS + j] - m);
      sc[q * SPS + j] = e;
      s += e;
    }
    for (int o = 16; o > 0; o >>= 1) s += __shfl_xor(s, o, 32);
    const float inv = 1.f / s;
    for (int j = lane; j < KSZ; j += 32) sc[q * SPS + j] *= inv;
  }
  __syncthreads();

  // ---------- Phase 4: out(16x64) = P(16x512) @ kh(512x64) -----------------
  if (wv < 4) {
    const int nt = wv;                                  // d-tile
    v8f c = {0.f,0.f,0.f,0.f,0.f,0.f,0.f,0.f};
    const float* xb = x + (size_t)b * KSZ * KDZ + h * DH + nt * 16 + ln;
    for (int kk = 0; kk < KSZ / 4; ++kk) {
      const int kb = kk * 4 + hl * 2;
      v2f av, bv;
      av.x = sc[ln * SPS + kb];
      av.y = sc[ln * SPS + kb + 1];
      bv.x = xb[(size_t)kb * KDZ];
      bv.y = xb[(size_t)(kb + 1) * KDZ];
      c = wmma4(av, bv, c);
    }
    #pragma unroll
    for (int r = 0; r < 8; ++r)
      out[((size_t)b * QSZ + q0 + r + 8 * hl) * KDZ + h * DH + nt * 16 + ln] = c[r];
  }
}

extern "C" void kernel_launch(void* const* d_in, const int* in_sizes, int n_in,
                              void* d_out, int out_size, void* d_ws, size_t ws_size,
                              hipStream_t stream) {
  (void)in_sizes; (void)n_in; (void)out_size; (void)d_ws; (void)ws_size;
  const float* x     = (const float*)d_in[0];
  const float* query = (const float*)d_in[1];
  const float* W     = (const float*)d_in[2];
  const float* bias  = (const float*)d_in[3];
  const float* w2    = (const float*)d_in[4];
  const unsigned char* mask = (const unsigned char*)d_in[5];
  float* out = (float*)d_out;

  dim3 grid(BB * HH * (QSZ / QT));   // 8*8*4 = 256 workgroups
  dim3 block(256);                   // 8 waves (wave32)
  addatten_fused<<<grid, block, 0, stream>>>(x, query, W, bias, w2, mask, out);
}